// BipolarMorphological2D_Torch_20409684591419
// MI455X (gfx1250) — compile-verified
//
#include <hip/hip_runtime.h>
#include <math.h>

// Problem constants (fixed by the reference)
#define BATCH 8
#define CIN   32
#define HH    32
#define WW    32
#define OUTC  64
#define FH    3
#define FW    3
#define HO    30
#define WO    30
#define PTOT  (FH*FW*CIN)      // 288
#define ESZ   (PTOT*OUTC)      // 18432 floats per kernel table
#define SHIFT 0.1f

// Tile config: one workgroup = 5 output pixels (one wave each) x 64 channels
#define QPIX  5
#define NTHR  (QPIX*32)        // 160 threads = 5 waves
#define TCOLS (QPIX + FW - 1)  // 7 halo columns
#define STAGE (CIN*FH*TCOLS)   // 672 staged floats

// ---- CDNA5 feature detection (compile-only safe fallbacks) ----
#if defined(__has_builtin)
#  if __has_builtin(__builtin_amdgcn_global_load_async_to_lds_b32)
#    define HAVE_ASYNC_LDS 1
#  endif
#  if __has_builtin(__builtin_amdgcn_s_wait_asynccnt)
#    define HAVE_WAIT_ASYNC 1
#  endif
#  if __has_builtin(__builtin_amdgcn_tensor_load_to_lds) && __has_builtin(__builtin_amdgcn_s_wait_tensorcnt)
#    define HAVE_TDM 1
#  endif
#endif
#ifndef HAVE_ASYNC_LDS
#  define HAVE_ASYNC_LDS 0
#endif
#ifndef HAVE_WAIT_ASYNC
#  define HAVE_WAIT_ASYNC 0
#endif
#ifndef HAVE_TDM
#  define HAVE_TDM 0
#endif

// ROCm 7.2 clang-22: 5-arg tensor builtin.  amdgpu-toolchain (therock-10.0
// headers present): 6-arg form with an extra i32x8 group.
#if HAVE_TDM && defined(__has_include)
#  if __has_include(<hip/amd_detail/amd_gfx1250_TDM.h>)
#    define TDM_ARITY6 1
#  endif
#endif
#ifndef TDM_ARITY6
#  define TDM_ARITY6 0
#endif

#define AS1 __attribute__((address_space(1)))
#define AS3 __attribute__((address_space(3)))

typedef unsigned int u32x4 __attribute__((ext_vector_type(4)));
typedef int          i32x4 __attribute__((ext_vector_type(4)));
typedef int          i32x8 __attribute__((ext_vector_type(8)));

// Kernel 1: e = exp(k) for both kernel tables (hoists all transcendentals).
__global__ void prep_exp_kernel(const float* __restrict__ k1,
                                const float* __restrict__ k2,
                                float* __restrict__ e1,
                                float* __restrict__ e2) {
    int i = blockIdx.x * blockDim.x + threadIdx.x;
    if (i < ESZ) {
        e1[i] = expf(k1[i]);
        e2[i] = expf(k2[i]);
    }
}

// Kernel 2: tropical (max,*) conv over 4 branches.
// out[b,o,ho,wo] = max_p(xp*e1) - max_p(xp*e2) - max_p(xn*e1) + max_p(xn*e2) + bias[o]
// with xp = max(x,0.1), xn = max(-x,0.1)  (exactly exp() of the reference's log-domain form).
__global__ void __launch_bounds__(NTHR)
morpho_main_kernel(const float* __restrict__ xg,
                   const float* __restrict__ e1f,
                   const float* __restrict__ e2f,
                   const float* __restrict__ bias,
                   float* __restrict__ out) {
    __shared__ float s_raw[STAGE];
    __shared__ float s_xp[STAGE];
    __shared__ float s_xn[STAGE];

    const int t    = threadIdx.x;
    const int lane = t & 31;        // channel pair index: owns o = 2*lane, 2*lane+1
    const int q    = t >> 5;        // 0..4 : pixel within tile (one wave per pixel)
    const int wt   = blockIdx.x;    // 0..5
    const int ho   = blockIdx.y;    // 0..29
    const int b    = blockIdx.z;    // 0..7
    const int wo0  = wt * QPIX;

    // ---- Phase 1: stage the 3x7x32 input halo tile into LDS ----
    // LDS layout: idx = c*21 + fi*7 + col  (X fastest, then Y, then Z — TDM tile order)
#if HAVE_TDM
    if (q == 0) {
        // Tile start address in global memory: x[b, 0, ho, wo0]
        const unsigned long long gaddr =
            (unsigned long long)(uintptr_t)(xg + (((b * CIN) * HH + ho) * WW + wo0));
        const unsigned ldsaddr = (unsigned)(uintptr_t)(AS3 float*)s_raw;

        // ---- D# group 0 (ISA §8.3): count=1, lds_addr, global_addr, type=2 ----
        u32x4 g0;
        g0[0] = 1u;                                             // count=1, user desc
        g0[1] = ldsaddr;                                        // lds_addr (bytes)
        g0[2] = (unsigned)(gaddr & 0xffffffffu);                // global_addr[31:0]
        g0[3] = (unsigned)((gaddr >> 32) & 0x01ffffffu)         // global_addr[56:32]
              | (2u << 30);                                     // type = 2 ("image")

        // ---- D# group 1 (ISA §8.4) ----
        i32x8 g1;
        g1[0] = (2 << 16);                  // workgroup_mask=0, data_size=2 (4 bytes)
        g1[1] = (WW & 0xffff) << 16;        // tensor_dim0[15:0]  (=32) @bits 63:48
        g1[2] = (HH & 0xffff) << 16;        // tensor_dim0 hi=0; tensor_dim1[15:0] (=32) @95:80
        g1[3] = (TCOLS << 16);              // tensor_dim1 hi=0; tile_dim0 = 7 @127:112
        g1[4] = (CIN << 16) | FH;           // tile_dim1 = 3 @143:128; tile_dim2 = 32 @159:144
        g1[5] = WW;                         // tensor_dim0_stride = 32 @191:160
        g1[6] = (HH * WW) << 16;            // stride hi=0; tensor_dim1_stride lo = 1024 @223:208
        g1[7] = 0;                          // tensor_dim1_stride hi

        // ---- D# group 2: tensor_dim2 = 32 channels; 3D tile so dim3 unused ----
        i32x4 g2;
        g2[0] = CIN;                        // tensor_dim2
        g2[1] = 0;                          // tensor_dim3
        g2[2] = 0;                          // tensor_dim2_stride lo (unused for 3D tile)
        g2[3] = 0;                          // stride hi / tile_dim3 = 0

        i32x4 g3 = {0, 0, 0, 0};            // group 3 unused (no dim4)

#if TDM_ARITY6
        i32x8 gx = {0, 0, 0, 0, 0, 0, 0, 0};
        __builtin_amdgcn_tensor_load_to_lds(g0, g1, g2, g3, gx, 0);
#else
        __builtin_amdgcn_tensor_load_to_lds(g0, g1, g2, g3, 0);
#endif
        __builtin_amdgcn_s_wait_tensorcnt(0);
    }
#else
    for (int idx = t; idx < STAGE; idx += NTHR) {
        const int col = idx % TCOLS;
        const int rem = idx / TCOLS;
        const int fi  = rem % FH;
        const int c   = rem / FH;
        const int goff = ((b * CIN + c) * HH + (ho + fi)) * WW + (wo0 + col);
#if HAVE_ASYNC_LDS
        __builtin_amdgcn_global_load_async_to_lds_b32(
            (AS1 int*)(xg + goff), (AS3 int*)(s_raw + idx), 0, 0);
#else
        s_raw[idx] = xg[goff];
#endif
    }
#if HAVE_ASYNC_LDS
#  if HAVE_WAIT_ASYNC
    __builtin_amdgcn_s_wait_asynccnt(0);
#  else
    asm volatile("s_wait_asynccnt 0" ::: "memory");
#  endif
#endif
#endif // HAVE_TDM
    __syncthreads();

    // ---- Phase 2: materialize both clamped branches in LDS ----
    for (int idx = t; idx < STAGE; idx += NTHR) {
        const float v = s_raw[idx];
        s_xp[idx] = fmaxf(v, SHIFT);
        s_xn[idx] = fmaxf(-v, SHIFT);
    }
    __syncthreads();

    // ---- Phase 3: max-times inner loop (pure v_mul / v_max_num, dual-issue friendly) ----
    const float2* __restrict__ e1v = (const float2*)e1f;  // [p][o/2] pairs
    const float2* __restrict__ e2v = (const float2*)e2f;

    float m11a = 0.f, m11b = 0.f, m12a = 0.f, m12b = 0.f;
    float m21a = 0.f, m21b = 0.f, m22a = 0.f, m22b = 0.f;

    for (int fi = 0; fi < FH; ++fi) {
        for (int fj = 0; fj < FW; ++fj) {
            const int colbase = fi * TCOLS + (q + fj);   // uniform per wave -> LDS broadcast
            const int pbase   = (fi * FW + fj) * CIN;
            // Warm the next e-block in L2/WGP$ (global_prefetch_b8 on gfx1250)
            const int pnext = pbase + CIN;
            if (pnext < PTOT) {
                __builtin_prefetch((const void*)(e1v + pnext * (OUTC/2) + lane), 0, 3);
                __builtin_prefetch((const void*)(e2v + pnext * (OUTC/2) + lane), 0, 3);
            }
#pragma unroll 8
            for (int c = 0; c < CIN; ++c) {
                const float xpv = s_xp[c * (FH*TCOLS) + colbase];
                const float xnv = s_xn[c * (FH*TCOLS) + colbase];
                const float2 a = e1v[(pbase + c) * (OUTC/2) + lane];
                const float2 d = e2v[(pbase + c) * (OUTC/2) + lane];
                m11a = fmaxf(m11a, xpv * a.x);  m11b = fmaxf(m11b, xpv * a.y);
                m12a = fmaxf(m12a, xpv * d.x);  m12b = fmaxf(m12b, xpv * d.y);
                m21a = fmaxf(m21a, xnv * a.x);  m21b = fmaxf(m21b, xnv * a.y);
                m22a = fmaxf(m22a, xnv * d.x);  m22b = fmaxf(m22b, xnv * d.y);
            }
        }
    }

    // ---- Output: [B, O, HO, WO] ----
    const int o0 = lane << 1;
    const int wo = wo0 + q;
    const float r0 = m11a - m12a - m21a + m22a + bias[o0];
    const float r1 = m11b - m12b - m21b + m22b + bias[o0 + 1];
    out[((b * OUTC + o0    ) * HO + ho) * WO + wo] = r0;
    out[((b * OUTC + o0 + 1) * HO + ho) * WO + wo] = r1;
}

extern "C" void kernel_launch(void* const* d_in, const int* in_sizes, int n_in,
                              void* d_out, int out_size, void* d_ws, size_t ws_size,
                              hipStream_t stream) {
    const float* x    = (const float*)d_in[0];
    const float* k1   = (const float*)d_in[1];
    const float* k2   = (const float*)d_in[2];
    const float* bias = (const float*)d_in[3];

    float* e1 = (float*)d_ws;          // 18432 floats
    float* e2 = e1 + ESZ;              // 18432 floats  (total 147 KB of ws)

    prep_exp_kernel<<<dim3((ESZ + 255) / 256), dim3(256), 0, stream>>>(k1, k2, e1, e2);

    dim3 grid(WO / QPIX, HO, BATCH);   // (6, 30, 8) = 1440 workgroups
    morpho_main_kernel<<<grid, dim3(NTHR), 0, stream>>>(x, e1, e2, bias, (float*)d_out);
}